// calculatorLocalEnergy_16097537425873
// MI455X (gfx1250) — compile-verified
//
#include <hip/hip_runtime.h>
#include <hip/hip_bf16.h>
#include <math.h>
#include <stdint.h>

// Problem constants
#define NA 64
#define NH 1024
#define NB 512

#define GAUSS_CONST (-3.9894228040143270f)   // V0 / (sqrt(2*pi)*sigma0), V0=-5, sigma0=0.5
#define INV_2SIG2   (2.0f)                   // 1 / (2*sigma0^2)

// Hessian GEMM blocking
#define COLS  512      // columns of W2 per block
#define KCH   64       // K chunk staged in LDS
#define NCHUNK (NH / KCH)

typedef float v2f __attribute__((ext_vector_type(2)));
typedef float v8f __attribute__((ext_vector_type(8)));

// ---------------------------------------------------------------------------
// Block-wide sum reduction (blockDim.x == 256), returns total to all threads.
// ---------------------------------------------------------------------------
__device__ __forceinline__ float block_reduce_256(float v, float* red) {
    const int t = threadIdx.x;
    red[t] = v;
    __syncthreads();
    #pragma unroll
    for (int s = 128; s > 0; s >>= 1) {
        if (t < s) red[t] += red[t + s];
        __syncthreads();
    }
    float r = red[0];
    __syncthreads();
    return r;
}

// ---------------------------------------------------------------------------
// Q[i] = sum_a W1[a,i]^2   (sample independent)  — grid 4 x 256
// ---------------------------------------------------------------------------
__global__ __launch_bounds__(256) void q_kernel(const float* __restrict__ W1,
                                                float* __restrict__ Q) {
    int i = blockIdx.x * 256 + threadIdx.x;  // 0..1023
    float q = 0.f;
    #pragma unroll 8
    for (int a = 0; a < NA; ++a) {
        float w = W1[a * NH + i];
        q = fmaf(w, w, q);
    }
    Q[i] = q;
}

// ---------------------------------------------------------------------------
// Per-sample forward + analytic backward (everything except the big GEMM).
// One block (256 threads) per sample.
// Writes: t1g[s,H], cg[s,H], baseg[s] = pot + gint - 0.5*(T2 + |g|^2)
// ---------------------------------------------------------------------------
__global__ __launch_bounds__(256) void fwd_kernel(const float* __restrict__ x,
                                                  const float* __restrict__ W1,
                                                  const float* __restrict__ b1,
                                                  const float* __restrict__ W2,
                                                  const float* __restrict__ b2,
                                                  const float* __restrict__ W3,
                                                  const float* __restrict__ Q,
                                                  float* __restrict__ t1g,
                                                  float* __restrict__ cg,
                                                  float* __restrict__ baseg) {
    const int s = blockIdx.x;
    const int t = threadIdx.x;

    __shared__ float xs[NA];
    __shared__ float h1s[NH];
    __shared__ float s2s[NH];
    __shared__ float s1s[NH];
    __shared__ float red[256];

    if (t < NA) xs[t] = x[s * NA + t];
    __syncthreads();

    // ---- layer 1: z1 = x@W1 + b1, h1 = tanh, t1 = 1-h1^2 ----
    #pragma unroll
    for (int k = 0; k < 4; ++k) {
        const int i = t + 256 * k;
        float acc = b1[i];
        #pragma unroll 8
        for (int a = 0; a < NA; ++a)
            acc = fmaf(xs[a], W1[a * NH + i], acc);
        float h = tanhf(acc);
        h1s[i] = h;
        t1g[s * NH + i] = 1.f - h * h;
    }
    __syncthreads();

    // ---- layer 2: z2 = h1@W2 + b2; s2 = t2*W3; c = -2*h2*t2*W3 ----
    #pragma unroll
    for (int k = 0; k < 4; ++k) {
        const int j = t + 256 * k;
        float acc = b2[j];
        for (int i = 0; i < NH; ++i)
            acc = fmaf(h1s[i], W2[i * NH + j], acc);
        float h2 = tanhf(acc);
        float t2 = 1.f - h2 * h2;
        float w3 = W3[j];
        s2s[j] = t2 * w3;
        cg[s * NH + j] = -2.f * h2 * t2 * w3;
    }
    __syncthreads();

    // ---- d1 = W2 @ s2 ; s1 = t1*d1 ; T2 = sum_i(-2 h1 t1 Q d1) ----
    float T2p = 0.f;
    #pragma unroll
    for (int k = 0; k < 4; ++k) {
        const int i = t + 256 * k;
        const float* wrow = W2 + (size_t)i * NH;
        float d = 0.f;
        for (int j = 0; j < NH; ++j)
            d = fmaf(wrow[j], s2s[j], d);
        float h1 = h1s[i];
        float t1 = 1.f - h1 * h1;
        s1s[i] = t1 * d;
        T2p = fmaf(-2.f * h1 * t1 * Q[i], d, T2p);
    }
    float T2 = block_reduce_256(T2p, red);

    // ---- g = W1 @ s1 ; |g|^2 ----
    float gsqp = 0.f;
    if (t < NA) {
        const float* wrow = W1 + (size_t)t * NH;
        float ga = 0.f;
        for (int i = 0; i < NH; ++i)
            ga = fmaf(wrow[i], s1s[i], ga);
        gsqp = ga * ga;
    }
    float gsq = block_reduce_256(gsqp, red);

    // ---- potential + pairwise Gaussian interaction ----
    float pp = 0.f;
    if (t < NA) pp = 0.5f * xs[t] * xs[t];
    float gp = 0.f;
    for (int e = t; e < NA * NA; e += 256) {
        int a = e >> 6, b = e & 63;
        if (b > a) {
            float d = xs[a] - xs[b];
            gp += expf(-(d * d) * INV_2SIG2);
        }
    }
    float pe = block_reduce_256(pp + GAUSS_CONST * gp, red);

    if (t == 0) baseg[s] = pe - 0.5f * (T2 + gsq);
}

// ---------------------------------------------------------------------------
// Hessian-trace GEMM term via f32 WMMA, async-LDS double-buffered:
//   Z[a,j] = sum_i (W1[a,i]*t1[i]) * W2[i,j]   (M=64, K=1024, N=COLS per block)
//   T1part[s,jb] = sum_{j in block} c[j] * sum_a Z[a,j]^2
// Grid: (NB, NH/COLS); block: 256 threads = 8 waves.
// Wave w owns ALL 4 row-tiles x 4 col-tiles (col-tiles 4w..4w+3):
//   16 WMMAs per K=4 step for 8 ds_load_2addr.
// W2 slabs stream into the spare buffer via GLOBAL_LOAD_ASYNC_TO_LDS_B128
// (ASYNCcnt) while the current chunk computes; A_eff (needs the t1 multiply)
// is staged synchronously alongside. Handoff: s_wait_asynccnt 0 + barrier.
// ---------------------------------------------------------------------------
__global__ __launch_bounds__(256) void hess_wmma_kernel(const float* __restrict__ W1,
                                                        const float* __restrict__ W2,
                                                        const float* __restrict__ t1g,
                                                        const float* __restrict__ cg,
                                                        float* __restrict__ T1part) {
    const int s  = blockIdx.x;
    const int jb = blockIdx.y;
    const int j0 = jb * COLS;
    const int tid  = threadIdx.x;
    const int wave = tid >> 5;
    const int lane = tid & 31;

    __shared__ float As[2][64 * 65];      // A_eff chunks (64 x KCH), padded stride
    __shared__ float Bs[2][KCH * COLS];   // W2 chunks (KCH x COLS) = 2 x 128 KB
    __shared__ float red[256];

    const int nlane = lane & 15;          // N (and M) index within a 16x16 tile
    const int klane = (lane >> 4) * 2;    // K sub-offset per ISA f32 16x4 layout
    const int ct0   = wave * 4;           // first of this wave's 4 col-tiles (of 32)

    const float* t1row = t1g + (size_t)s * NH;

    v8f acc[4][4];                        // [rowTile][colTile]
    #pragma unroll
    for (int rt = 0; rt < 4; ++rt)
        #pragma unroll
        for (int ct = 0; ct < 4; ++ct)
            acc[rt][ct] = (v8f){};

    // ---- staging helpers -------------------------------------------------
    // Async W2 slab: 64 x COLS floats = 8192 b128 transfers, 32 per thread.
    // LDS destination address = low 32 bits of the flat shared-mem address
    // (ISA aperture rule).
    auto stage_b_async = [&](int k0, int buf) {
        const uint32_t lbase = (uint32_t)(uintptr_t)(&Bs[buf][0]);
        #pragma unroll
        for (int r = 0; r < 32; ++r) {
            int e   = tid + 256 * r;       // float4 index 0..8191
            int row = e >> 7;
            int c4  = (e & 127) * 4;
            const float* gp = &W2[(size_t)(k0 + row) * NH + j0 + c4];
            uint32_t loff = lbase + (uint32_t)((row * COLS + c4) * 4);
            asm volatile("global_load_async_to_lds_b128 %0, %1, off"
                         :: "v"(loff), "v"(gp)
                         : "memory");
        }
    };
    // A_eff[a, k0+i] = W1[a, k0+i] * t1[k0+i]  (64 x 64, synchronous: needs VALU)
    auto stage_a = [&](int k0, int buf) {
        #pragma unroll
        for (int r = 0; r < 16; ++r) {
            int e = tid + 256 * r;
            int a = e >> 6, i = e & 63;
            As[buf][a * 65 + i] = W1[a * NH + k0 + i] * t1row[k0 + i];
        }
    };

    // ---- prologue: fill buffer 0 ----------------------------------------
    stage_b_async(0, 0);
    stage_a(0, 0);
    asm volatile("s_wait_asynccnt 0x0" ::: "memory");
    __syncthreads();

    for (int c = 0; c < NCHUNK; ++c) {
        const int cur = c & 1;

        // prefetch next chunk into the spare buffer while we compute
        if (c + 1 < NCHUNK) {
            stage_b_async((c + 1) * KCH, cur ^ 1);
            stage_a((c + 1) * KCH, cur ^ 1);
        }

        const float* Ab = &As[cur][0];
        const float* Bb = &Bs[cur][0];

        for (int kk = 0; kk < KCH; kk += 4) {
            const int kidx = kk + klane;

            v2f av[4];
            #pragma unroll
            for (int rt = 0; rt < 4; ++rt) {
                const int arow = rt * 16 + nlane;
                av[rt].x = Ab[arow * 65 + kidx];
                av[rt].y = Ab[arow * 65 + kidx + 1];
            }
            v2f bv[4];
            #pragma unroll
            for (int ct = 0; ct < 4; ++ct) {
                const int col = (ct0 + ct) * 16 + nlane;
                bv[ct].x = Bb[kidx * COLS + col];
                bv[ct].y = Bb[(kidx + 1) * COLS + col];
            }
            #pragma unroll
            for (int rt = 0; rt < 4; ++rt)
                #pragma unroll
                for (int ct = 0; ct < 4; ++ct)
                    acc[rt][ct] = __builtin_amdgcn_wmma_f32_16x16x4_f32(
                        false, av[rt], false, bv[ct], (short)0, acc[rt][ct],
                        false, false);
        }

        // own async transfers landed; barrier -> everyone done with old buffer
        asm volatile("s_wait_asynccnt 0x0" ::: "memory");
        __syncthreads();
    }

    // weighted column sums of squares:  p += c[j] * sum_a Z[a,j]^2
    const float* crow = cg + (size_t)s * NH;
    float pv = 0.f;
    #pragma unroll
    for (int ct = 0; ct < 4; ++ct) {
        const int j = j0 + (ct0 + ct) * 16 + nlane;
        const float cj = crow[j];
        float ss = 0.f;
        #pragma unroll
        for (int rt = 0; rt < 4; ++rt) {
            #pragma unroll
            for (int r = 0; r < 8; ++r) {
                float v = acc[rt][ct][r];
                ss = fmaf(v, v, ss);
            }
        }
        pv = fmaf(cj, ss, pv);
    }

    float tot = block_reduce_256(pv, red);
    if (tid == 0) T1part[s * (NH / COLS) + jb] = tot;
}

// ---------------------------------------------------------------------------
// out[s] = base[s] - 0.5 * sum_jb T1part[s,jb]
// ---------------------------------------------------------------------------
__global__ __launch_bounds__(256) void finish_kernel(const float* __restrict__ baseg,
                                                     const float* __restrict__ T1part,
                                                     float* __restrict__ out) {
    int s = blockIdx.x * 256 + threadIdx.x;
    if (s < NB) {
        float t1 = 0.f;
        #pragma unroll
        for (int jb = 0; jb < NH / COLS; ++jb) t1 += T1part[s * (NH / COLS) + jb];
        out[s] = baseg[s] - 0.5f * t1;
    }
}

// ---------------------------------------------------------------------------
extern "C" void kernel_launch(void* const* d_in, const int* in_sizes, int n_in,
                              void* d_out, int out_size, void* d_ws, size_t ws_size,
                              hipStream_t stream) {
    const float* x  = (const float*)d_in[0];  // [B, A]
    const float* W1 = (const float*)d_in[1];  // [A, H]
    const float* b1 = (const float*)d_in[2];  // [H]
    const float* W2 = (const float*)d_in[3];  // [H, H]
    const float* b2 = (const float*)d_in[4];  // [H]
    const float* W3 = (const float*)d_in[5];  // [H, 1]
    // d_in[6] = b3 — does not affect derivatives or the energy.
    float* out = (float*)d_out;

    // Workspace layout (floats)
    float* ws     = (float*)d_ws;
    float* Q      = ws;                        // 1024
    float* t1g    = Q + NH;                    // B*H
    float* cg     = t1g + (size_t)NB * NH;     // B*H
    float* baseg  = cg + (size_t)NB * NH;      // B
    float* T1part = baseg + NB;                // B * (NH/COLS)
    size_t need = (size_t)(NH + 2 * NB * NH + NB + NB * (NH / COLS)) * sizeof(float);
    if (ws_size < need) return;

    q_kernel<<<dim3(NH / 256), dim3(256), 0, stream>>>(W1, Q);
    fwd_kernel<<<dim3(NB), dim3(256), 0, stream>>>(x, W1, b1, W2, b2, W3, Q,
                                                   t1g, cg, baseg);
    hess_wmma_kernel<<<dim3(NB, NH / COLS), dim3(256), 0, stream>>>(W1, W2, t1g, cg, T1part);
    finish_kernel<<<dim3((NB + 255) / 256), dim3(256), 0, stream>>>(baseg, T1part, out);
}